// Encoder_82308753260928
// MI455X (gfx1250) — compile-verified
//
#include <hip/hip_runtime.h>
#include <hip/hip_bf16.h>

#define N_NODES 50000
#define N_EDGES 800000
#define HDIM    128
#define LAYERS  3
#define NGRAPH  64
#define BN_EPS  1e-5f

typedef __attribute__((ext_vector_type(2))) float v2f;
typedef __attribute__((ext_vector_type(8))) float v8f;

// ---------------------------------------------------------------- degree
__global__ __launch_bounds__(256) void k_deg_init(float* __restrict__ deg) {
    int n = blockIdx.x * blockDim.x + threadIdx.x;
    if (n < N_NODES) deg[n] = 1.0f;   // self-loop
}

__global__ __launch_bounds__(256) void k_deg_accum(const int* __restrict__ col,
                                                   float* __restrict__ deg) {
    int e = blockIdx.x * blockDim.x + threadIdx.x;
    if (e < N_EDGES) atomicAdd(&deg[col[e]], 1.0f);
}

__global__ __launch_bounds__(256) void k_dinv(float* __restrict__ deg) {
    int n = blockIdx.x * blockDim.x + threadIdx.x;
    if (n < N_NODES) deg[n] = rsqrtf(deg[n]);   // deg >= 1 always
}

// ---------------------------------------------------------------- embedding
__global__ __launch_bounds__(256) void k_embed(const int* __restrict__ x,
                                               const float* __restrict__ embed,
                                               float* __restrict__ h) {
    size_t i = (size_t)blockIdx.x * blockDim.x + threadIdx.x;
    if (i >= (size_t)N_NODES * HDIM) return;
    int n = (int)(i >> 7);
    int c = (int)(i & 127);
    h[i] = embed[(size_t)x[n] * HDIM + c];
}

// ---------------------------------------------------------------- GEMM (WMMA f32)
// m = h @ W   [N,128] x [128,128]
// block = 256 threads = 8 waves; block b handles rows [16b,16b+16); wave w -> cols [16w,16w+16)
// f32 WMMA 16x16x4 VGPR striping (ISA 7.12.2):
//   A 16x4:  lanes 0-15 hold M=lane,   v0=K0 v1=K1 ; lanes 16-31 hold M=lane-16, v0=K2 v1=K3
//   B 4x16:  lanes 0-15 hold N=lane,   v0=K0 v1=K1 ; lanes 16-31 hold N=lane-16, v0=K2 v1=K3
//   C 16x16: VGPR r: lanes 0-15 = row r, lanes 16-31 = row r+8
__global__ __launch_bounds__(256) void k_gemm_wmma(const float* __restrict__ h,
                                                   const float* __restrict__ W,
                                                   float* __restrict__ m) {
    const int wave = threadIdx.x >> 5;
    const int lane = threadIdx.x & 31;
    const int l16  = lane & 15;
    const int hi   = lane >> 4;           // 0: K{0,1}  1: K{2,3}
    const int row0 = blockIdx.x * 16;
    const int col0 = wave * 16;

    const float* A = h + (size_t)row0 * HDIM;
    v8f acc = {};
    #pragma unroll 4
    for (int k = 0; k < HDIM; k += 4) {
        const int ka = k + hi * 2;
        v2f a, b;
        a.x = A[(size_t)l16 * HDIM + ka + 0];
        a.y = A[(size_t)l16 * HDIM + ka + 1];
        b.x = W[(size_t)(ka + 0) * HDIM + col0 + l16];
        b.y = W[(size_t)(ka + 1) * HDIM + col0 + l16];
        acc = __builtin_amdgcn_wmma_f32_16x16x4_f32(
            /*neg_a=*/false, a, /*neg_b=*/false, b,
            /*c_mod=*/(short)0, acc, /*reuse_a=*/false, /*reuse_b=*/false);
    }
    float* O = m + (size_t)row0 * HDIM + col0;
    #pragma unroll
    for (int r = 0; r < 8; ++r)
        O[(size_t)(r + hi * 8) * HDIM + l16] = acc[r];
}

// ---------------------------------------------------------------- aggregation
// agg[n][c] = b[c] + dinv[n]^2 * m[n][c]   (bias + self-loop term)
__global__ __launch_bounds__(256) void k_agg_init(const float* __restrict__ m,
                                                  const float* __restrict__ dinv,
                                                  const float* __restrict__ bias,
                                                  float* __restrict__ agg) {
    size_t i = (size_t)blockIdx.x * blockDim.x + threadIdx.x;
    if (i >= (size_t)N_NODES * HDIM) return;
    int n = (int)(i >> 7);
    int c = (int)(i & 127);
    float d = dinv[n];
    agg[i] = bias[c] + d * d * m[i];
}

// one wave per edge; lane handles 4 consecutive channels (float4 gather)
__global__ __launch_bounds__(256) void k_scatter(const int* __restrict__ row,
                                                 const int* __restrict__ col,
                                                 const float* __restrict__ dinv,
                                                 const float* __restrict__ m,
                                                 float* __restrict__ agg) {
    long long tid = (long long)blockIdx.x * blockDim.x + threadIdx.x;
    int e = (int)(tid >> 5);
    if (e >= N_EDGES) return;
    int lane = (int)(tid & 31);
    int r = row[e], c = col[e];
    float nrm = dinv[r] * dinv[c];
    const float4* ms = (const float4*)(m + (size_t)r * HDIM);
    float4 v = ms[lane];
    float* dst = agg + (size_t)c * HDIM + lane * 4;
    atomicAdd(dst + 0, nrm * v.x);
    atomicAdd(dst + 1, nrm * v.y);
    atomicAdd(dst + 2, nrm * v.z);
    atomicAdd(dst + 3, nrm * v.w);
}

// ---------------------------------------------------------------- batchnorm
__global__ __launch_bounds__(128) void k_stats_zero(float* __restrict__ ssum,
                                                    float* __restrict__ ssq) {
    ssum[threadIdx.x] = 0.0f;
    ssq[threadIdx.x]  = 0.0f;
}

// 128 threads = 1 channel each; block b covers a node chunk; coalesced loads
__global__ __launch_bounds__(128) void k_bn_stats(const float* __restrict__ agg,
                                                  float* __restrict__ ssum,
                                                  float* __restrict__ ssq) {
    const int c = threadIdx.x;
    const int chunk = (N_NODES + gridDim.x - 1) / gridDim.x;
    int n0 = blockIdx.x * chunk;
    int n1 = n0 + chunk; if (n1 > N_NODES) n1 = N_NODES;
    float s = 0.0f, q = 0.0f;
    for (int n = n0; n < n1; ++n) {
        float a = agg[(size_t)n * HDIM + c];
        a = a > 0.0f ? a : 0.0f;        // ReLU before stats
        s += a;
        q += a * a;
    }
    atomicAdd(&ssum[c], s);
    atomicAdd(&ssq[c],  q);
}

__global__ __launch_bounds__(256) void k_bn_apply(const float* __restrict__ agg,
                                                  const float* __restrict__ ssum,
                                                  const float* __restrict__ ssq,
                                                  const float* __restrict__ gamma,
                                                  const float* __restrict__ beta,
                                                  float* __restrict__ h) {
    size_t i = (size_t)blockIdx.x * blockDim.x + threadIdx.x;
    if (i >= (size_t)N_NODES * HDIM) return;
    int c = (int)(i & 127);
    float a = agg[i];
    a = a > 0.0f ? a : 0.0f;
    float inv_n = 1.0f / (float)N_NODES;
    float mu  = ssum[c] * inv_n;
    float var = ssq[c] * inv_n - mu * mu;
    h[i] = gamma[c] * (a - mu) * rsqrtf(var + BN_EPS) + beta[c];
}

// ---------------------------------------------------------------- pooling
__global__ __launch_bounds__(256) void k_pool_zero(float* __restrict__ pool,
                                                   float* __restrict__ cnt) {
    int i = blockIdx.x * blockDim.x + threadIdx.x;
    if (i < NGRAPH * HDIM) pool[i] = 0.0f;
    if (i < NGRAPH) cnt[i] = 0.0f;
}

// one wave per node
__global__ __launch_bounds__(256) void k_pool_accum(const float* __restrict__ h,
                                                    const int* __restrict__ batch,
                                                    float* __restrict__ pool,
                                                    float* __restrict__ cnt) {
    long long tid = (long long)blockIdx.x * blockDim.x + threadIdx.x;
    int n = (int)(tid >> 5);
    if (n >= N_NODES) return;
    int lane = (int)(tid & 31);
    int g = batch[n];
    const float4* hs = (const float4*)(h + (size_t)n * HDIM);
    float4 v = hs[lane];
    float* dst = pool + (size_t)g * HDIM + lane * 4;
    atomicAdd(dst + 0, v.x);
    atomicAdd(dst + 1, v.y);
    atomicAdd(dst + 2, v.z);
    atomicAdd(dst + 3, v.w);
    if (lane == 0) atomicAdd(&cnt[g], 1.0f);
}

__global__ __launch_bounds__(256) void k_pool_final(const float* __restrict__ pool,
                                                    const float* __restrict__ cnt,
                                                    float* __restrict__ out) {
    int i = blockIdx.x * blockDim.x + threadIdx.x;
    if (i >= NGRAPH * HDIM) return;
    int g = i >> 7;
    float c = cnt[g];
    out[i] = pool[i] / (c > 1.0f ? c : 1.0f);
}

// ---------------------------------------------------------------- launch
extern "C" void kernel_launch(void* const* d_in, const int* in_sizes, int n_in,
                              void* d_out, int out_size, void* d_ws, size_t ws_size,
                              hipStream_t stream) {
    (void)in_sizes; (void)n_in; (void)out_size; (void)ws_size;

    const int*   x     = (const int*)d_in[0];
    const int*   erow  = (const int*)d_in[1];                 // edge_index[0]
    const int*   ecol  = ((const int*)d_in[1]) + N_EDGES;     // edge_index[1]
    const int*   batch = (const int*)d_in[2];
    const float* embed = (const float*)d_in[3];
    const float* W     = (const float*)d_in[4];               // [L,H,H]
    const float* bias  = (const float*)d_in[5];               // [L,H]
    const float* gamma = (const float*)d_in[6];               // [L,H]
    const float* beta  = (const float*)d_in[7];               // [L,H]
    float*       out   = (float*)d_out;                       // [G,H]

    // workspace layout (floats)
    float* ws   = (float*)d_ws;
    const size_t NH = (size_t)N_NODES * HDIM;
    float* h    = ws;                 // [N,H]
    float* m    = ws + NH;            // [N,H]
    float* agg  = ws + 2 * NH;        // [N,H]
    float* dinv = ws + 3 * NH;        // [N]   (deg, then rsqrt in place)
    float* ssum = dinv + N_NODES;     // [H]
    float* ssq  = ssum + HDIM;        // [H]
    float* pool = ssq + HDIM;         // [G,H]
    float* cnt  = pool + NGRAPH * HDIM; // [G]

    const int nh_blocks   = (int)((NH + 255) / 256);                 // 25000
    const int node_blocks = (N_NODES + 255) / 256;
    const int edge_blocks = (N_EDGES + 255) / 256;
    const int scat_blocks = (int)(((long long)N_EDGES * 32 + 255) / 256);
    const int pacc_blocks = (int)(((long long)N_NODES * 32 + 255) / 256);

    // degrees -> dinv
    k_deg_init <<<node_blocks, 256, 0, stream>>>(dinv);
    k_deg_accum<<<edge_blocks, 256, 0, stream>>>(ecol, dinv);
    k_dinv     <<<node_blocks, 256, 0, stream>>>(dinv);

    // embedding
    k_embed<<<nh_blocks, 256, 0, stream>>>(x, embed, h);

    // GCN layers
    for (int l = 0; l < LAYERS; ++l) {
        const float* Wl = W     + (size_t)l * HDIM * HDIM;
        const float* bl = bias  + (size_t)l * HDIM;
        const float* gl = gamma + (size_t)l * HDIM;
        const float* el = beta  + (size_t)l * HDIM;

        k_gemm_wmma<<<N_NODES / 16, 256, 0, stream>>>(h, Wl, m);
        k_agg_init <<<nh_blocks, 256, 0, stream>>>(m, dinv, bl, agg);
        k_scatter  <<<scat_blocks, 256, 0, stream>>>(erow, ecol, dinv, m, agg);
        k_stats_zero<<<1, 128, 0, stream>>>(ssum, ssq);
        k_bn_stats <<<256, 128, 0, stream>>>(agg, ssum, ssq);
        k_bn_apply <<<nh_blocks, 256, 0, stream>>>(agg, ssum, ssq, gl, el, h);
    }

    // global mean pool
    k_pool_zero <<<(NGRAPH * HDIM + 255) / 256, 256, 0, stream>>>(pool, cnt);
    k_pool_accum<<<pacc_blocks, 256, 0, stream>>>(h, batch, pool, cnt);
    k_pool_final<<<(NGRAPH * HDIM + 255) / 256, 256, 0, stream>>>(pool, cnt, out);
}